// DETRPostProcessor_70214125355530
// MI455X (gfx1250) — compile-verified
//
#include <hip/hip_runtime.h>
#include <math.h>

#define NUM_CLASSES 80
#define QUERIES     1000
#define TOTV        80000          // scores per image
#define TOTV4       20000          // float4 per image
#define TOPK        300
#define HBITS       14
#define HBINS       (1 << HBITS)   // 16384 bins
#define HSHIFT      (32 - HBITS)   // 18
#define CAP         1024           // candidate buffer (expected ~305 entries)
#define SORTN       1024
#define NTHR        1024
#define NCHUNK      ((TOTV4 + NTHR - 1) / NTHR)   // 20 chunks of <=1024 float4

// gfx1250 async global->LDS path (ASYNCcnt-tracked), guarded so compile never breaks.
#if defined(__has_builtin)
#  if __has_builtin(__builtin_amdgcn_global_load_async_to_lds_b128) && \
      __has_builtin(__builtin_amdgcn_s_wait_asynccnt)
#    define USE_ASYNC_LDS 1
#  endif
#endif

typedef int v4i_t __attribute__((vector_size(16)));   // b128 payload type per clang diagnostic
typedef __attribute__((address_space(1))) v4i_t* gp_v4i;   // global (AS1) v4i*
typedef __attribute__((address_space(3))) v4i_t* lp_v4i;   // LDS (AS3) v4i*

// Monotone map: float -> u32 such that float order == unsigned order.
__device__ __forceinline__ unsigned mono(float f) {
  unsigned b = __float_as_uint(f);
  return (b & 0x80000000u) ? ~b : (b | 0x80000000u);
}

__global__ __launch_bounds__(NTHR) void detr_post_kernel(
    const float*  __restrict__ logits,
    const float4* __restrict__ boxes4,
    const int*    __restrict__ osz,
    float*        __restrict__ out)
{
  __shared__ unsigned hist[HBINS];            // 64 KB
  __shared__ unsigned bsum[256];
  __shared__ unsigned buf_u[CAP];
  __shared__ unsigned buf_i[CAP];
  __shared__ unsigned long long skey[SORTN];  // 8 KB
  __shared__ unsigned sh_cnt, sh_p14;
#ifdef USE_ASYNC_LDS
  __shared__ float4 stage[2][NTHR];           // 32 KB double buffer
#endif

  const int tid = threadIdx.x;
  const int b   = blockIdx.x;
  const float4* lg4 = reinterpret_cast<const float4*>(logits + (size_t)b * TOTV);

  // ---- init ----
#pragma unroll
  for (int j = 0; j < HBINS / NTHR; ++j) hist[tid + j * NTHR] = 0u;
  if (tid == 0) sh_cnt = 0u;
  __syncthreads();

  // ---- pass 1 (HBM stream): 14-bit histogram ----
  for (int i = tid; i < TOTV4; i += NTHR) {
    __builtin_prefetch(lg4 + i + 4 * NTHR, 0, 0);   // global_prefetch_b8
    float4 v = lg4[i];
    atomicAdd(&hist[mono(v.x) >> HSHIFT], 1u);
    atomicAdd(&hist[mono(v.y) >> HSHIFT], 1u);
    atomicAdd(&hist[mono(v.z) >> HSHIFT], 1u);
    atomicAdd(&hist[mono(v.w) >> HSHIFT], 1u);
  }
  __syncthreads();

  // ---- hierarchical descending scan: find 14-bit bucket of the 300th value ----
  if (tid < 256) {
    unsigned s = 0;
    const int base = tid * 64;
#pragma unroll 8
    for (int j = 0; j < 64; ++j) s += hist[base + j];
    bsum[tid] = s;
  }
  __syncthreads();
  if (tid == 0) {
    unsigned cum = 0;
    int Bsel = 0;
    for (int t = 255; t >= 0; --t) {
      unsigned s = bsum[t];
      if (cum + s >= TOPK) { Bsel = t; break; }
      cum += s;
    }
    unsigned p14 = (unsigned)Bsel * 64u;
    for (int j = 63; j >= 0; --j) {
      unsigned c = hist[Bsel * 64 + j];
      if (cum + c >= TOPK) { p14 = (unsigned)(Bsel * 64 + j); break; }
      cum += c;
    }
    sh_p14 = p14;
  }
  __syncthreads();
  const unsigned p14 = sh_p14;

  // ---- pass 2 (L2-resident): collect all elements with top14 >= p14 (~305 expected) ----
#ifdef USE_ASYNC_LDS
  {
    // prologue: async-stage chunk 0
    {
      int rem = TOTV4; if (rem > NTHR) rem = NTHR;
      if (tid < rem)
        __builtin_amdgcn_global_load_async_to_lds_b128(
            (gp_v4i)(lg4 + tid), (lp_v4i)&stage[0][tid], 0, 0);
    }
    for (int c = 0; c < NCHUNK; ++c) {
      const int nxt = c + 1;
      if (nxt < NCHUNK) {
        const int nb = nxt * NTHR;
        int remn = TOTV4 - nb; if (remn > NTHR) remn = NTHR;
        if (tid < remn)
          __builtin_amdgcn_global_load_async_to_lds_b128(
              (gp_v4i)(lg4 + nb + tid), (lp_v4i)&stage[nxt & 1][tid], 0, 0);
        __builtin_amdgcn_s_wait_asynccnt(1);   // oldest (chunk c) complete; in-order
      } else {
        __builtin_amdgcn_s_wait_asynccnt(0);
      }
      __asm__ volatile("" ::: "memory");       // keep LDS reads after the wait
      const int base = c * NTHR;
      int rem = TOTV4 - base; if (rem > NTHR) rem = NTHR;
      if (tid < rem) {
        float4 v = stage[c & 1][tid];
        float c4[4] = {v.x, v.y, v.z, v.w};
#pragma unroll
        for (int cc = 0; cc < 4; ++cc) {
          unsigned u = mono(c4[cc]);
          if ((u >> HSHIFT) >= p14) {
            unsigned p = atomicAdd(&sh_cnt, 1u);
            if (p < CAP) { buf_u[p] = u; buf_i[p] = (unsigned)(4 * (base + tid) + cc); }
          }
        }
      }
    }
  }
#else
  for (int i = tid; i < TOTV4; i += NTHR) {
    float4 v = lg4[i];
    float c4[4] = {v.x, v.y, v.z, v.w};
#pragma unroll
    for (int cc = 0; cc < 4; ++cc) {
      unsigned u = mono(c4[cc]);
      if ((u >> HSHIFT) >= p14) {
        unsigned p = atomicAdd(&sh_cnt, 1u);
        if (p < CAP) { buf_u[p] = u; buf_i[p] = (unsigned)(4 * i + cc); }
      }
    }
  }
#endif
  __syncthreads();
  const unsigned C = (sh_cnt < CAP) ? sh_cnt : CAP;

  // ---- bitonic sort (descending) of 64-bit keys: (score_bits << 32) | ~index ----
  // Full-key sort implements jax.lax.top_k ordering incl. smaller-index-first ties.
  skey[tid] = (tid < (int)C)
      ? (((unsigned long long)buf_u[tid] << 32) |
         (unsigned long long)(0xFFFFFFFFu - buf_i[tid]))
      : 0ull;
  __syncthreads();
  for (unsigned k2 = 2; k2 <= SORTN; k2 <<= 1) {
    for (unsigned j = k2 >> 1; j > 0; j >>= 1) {
      unsigned ixj = (unsigned)tid ^ j;
      if (ixj > (unsigned)tid) {
        unsigned long long a = skey[tid], bb = skey[ixj];
        bool descSeg = ((tid & k2) == 0);
        if (descSeg ? (a < bb) : (a > bb)) { skey[tid] = bb; skey[ixj] = a; }
      }
      __syncthreads();
    }
  }

  // ---- decode top-300 + box transform + write ----
  const float sW = (float)osz[1];   // original_sizes[0][::-1] -> [w, h, w, h]
  const float sH = (float)osz[0];
  if (tid < TOPK) {
    size_t o = ((size_t)b * TOPK + tid) * 6;
    if (tid < (int)C) {
      unsigned long long key = skey[tid];
      unsigned u    = (unsigned)(key >> 32);
      unsigned idx  = 0xFFFFFFFFu - (unsigned)(key & 0xFFFFFFFFull);
      unsigned bits = (u & 0x80000000u) ? (u & 0x7FFFFFFFu) : ~u;
      float f = __uint_as_float(bits);
      float score = 1.0f / (1.0f + expf(-f));
      unsigned q   = idx / NUM_CLASSES;
      unsigned lab = idx % NUM_CLASSES;
      float4 bx = boxes4[(size_t)b * QUERIES + q];
      out[o + 0] = (float)lab;
      out[o + 1] = score;
      out[o + 2] = (bx.x - 0.5f * bx.z) * sW;
      out[o + 3] = (bx.y - 0.5f * bx.w) * sH;
      out[o + 4] = bx.z * sW;
      out[o + 5] = bx.w * sH;
    } else {
      for (int c = 0; c < 6; ++c) out[o + c] = 0.0f;
    }
  }
}

extern "C" void kernel_launch(void* const* d_in, const int* in_sizes, int n_in,
                              void* d_out, int out_size, void* d_ws, size_t ws_size,
                              hipStream_t stream) {
  (void)n_in; (void)out_size; (void)d_ws; (void)ws_size;
  const float*  logits = (const float*)d_in[0];
  const float4* boxes4 = (const float4*)d_in[1];
  const int*    osz    = (const int*)d_in[2];
  float*        out    = (float*)d_out;
  const int N = in_sizes[0] / TOTV;   // 256 images
  detr_post_kernel<<<N, NTHR, 0, stream>>>(logits, boxes4, osz, out);
}